// HalutMatmul_13546326851791
// MI455X (gfx1250) — compile-verified
//
#include <hip/hip_runtime.h>
#include <hip/hip_bf16.h>

typedef float v2f __attribute__((ext_vector_type(2)));
typedef float v8f __attribute__((ext_vector_type(8)));

#define NROWS 16384
#define DIM   512
#define CSUB  64
#define KQ    16
#define MOUT  512

// ---------------------------------------------------------------------------
// Kernel 1: encode. Block = 256 threads = 4 rows x 64 subspaces.
// Writes codesT[c*NROWS + n] (uint8), transposed for decode-side broadcast.
// ---------------------------------------------------------------------------
__global__ __launch_bounds__(256) void halut_encode(
    const float* __restrict__ I, const float* __restrict__ A,
    const float* __restrict__ T, unsigned char* __restrict__ codesT) {
  __shared__ unsigned char lcode[4][CSUB];
  const int t  = threadIdx.x;
  const int c  = t & 63;
  const int rn = t >> 6;                       // 0..3
  const int n  = blockIdx.x * 4 + rn;

  // 8 input features of this subspace (row is contiguous: fully coalesced)
  const float* Ip = I + (size_t)n * DIM + c * 8;
  float4 i0 = ((const float4*)Ip)[0];
  float4 i1 = ((const float4*)Ip)[1];
  float xi[8] = {i0.x, i0.y, i0.z, i0.w, i1.x, i1.y, i1.z, i1.w};

  // xt[d] = sum_s xi[s] * A[c][s][d]   (A: (C, 8, 4))
  const float* Ac = A + c * 32;
  float xt[4] = {0.f, 0.f, 0.f, 0.f};
#pragma unroll
  for (int s = 0; s < 8; ++s) {
#pragma unroll
    for (int d = 0; d < 4; ++d) xt[d] = fmaf(xi[s], Ac[s * 4 + d], xt[d]);
  }

  // node activations: tanh(xt[level(i)] - T[c*15+i])
  const int lvl[15] = {0, 1, 1, 2, 2, 2, 2, 3, 3, 3, 3, 3, 3, 3, 3};
  const float* Tc = T + c * 15;
  float tn[15];
#pragma unroll
  for (int i = 0; i < 15; ++i) tn[i] = tanhf(xt[lvl[i]] - Tc[i]);

  // leaf scores b_k = sum over root-to-leaf path of +/- tanh; argmax (first max)
  float best = -3.0e38f;
  int bk = 0;
#pragma unroll
  for (int k = 0; k < KQ; ++k) {
    int node = 0;
    float b = 0.f;
#pragma unroll
    for (int l = 0; l < 4; ++l) {
      int bit = (k >> (3 - l)) & 1;
      b += bit ? tn[node] : -tn[node];
      node = 2 * node + 1 + bit;
    }
    if (b > best) { best = b; bk = k; }
  }
  lcode[rn][c] = (unsigned char)bk;
  __syncthreads();

  // transpose store: one dword (4 rows) per subspace
  if (t < CSUB) {
    uchar4 v = make_uchar4(lcode[0][t], lcode[1][t], lcode[2][t], lcode[3][t]);
    *(uchar4*)(codesT + (size_t)t * NROWS + blockIdx.x * 4) = v;
  }
}

// ---------------------------------------------------------------------------
// Kernel 2: pack L (M, C, K) fp32 into the WMMA f32 16x16x4 B-operand layout:
// P[(c*8 + q*2 + half)*512 + m] = { Lr[c*16+4q+2h][m], Lr[c*16+4q+2h+1][m] }
// so a b64 load per lane yields exactly (VGPR0, VGPR1) of the B matrix.
// ---------------------------------------------------------------------------
__global__ __launch_bounds__(256) void halut_pack_lut(
    const float* __restrict__ L, v2f* __restrict__ P) {
  int tid  = blockIdx.x * 256 + threadIdx.x;   // 64*8*512 = 262144 total
  int m    = tid & 511;
  int rest = tid >> 9;                          // (c*8 + q*2 + half), 0..511
  int c    = rest >> 3;
  int qh   = rest & 7;
  int kk   = (qh >> 1) * 4 + (qh & 1) * 2;      // 4q + 2*half
  const float* src = L + (size_t)m * (CSUB * KQ) + c * KQ + kk;
  v2f v;
  v[0] = src[0];
  v[1] = src[1];
  P[(size_t)rest * MOUT + m] = v;               // coalesced write
}

// ---------------------------------------------------------------------------
// Kernel 3: decode via V_WMMA_F32_16X16X4_F32.
// Each wave: one 16-col m-tile x eight 16-row n-tiles (128 rows).
// out = onehot(E) @ Lr, accumulated in exact fp32.
// ---------------------------------------------------------------------------
__global__ __launch_bounds__(256) void halut_decode(
    const v2f* __restrict__ P, const unsigned char* __restrict__ codesT,
    float* __restrict__ out) {
  const int lane  = threadIdx.x & 31;
  const int wave  = blockIdx.x * 8 + (threadIdx.x >> 5);
  const int mTile = wave & 31;                 // 32 m-tiles of 16
  const int nGrp  = wave >> 5;                 // 128 n-groups of 128 rows
  const int m0    = mTile * 16;
  const int n0    = nGrp * 128;
  const int half  = lane >> 4;                 // 0: K even pair base, 1: +2
  const int l15   = lane & 15;

  v8f acc[8];
#pragma unroll
  for (int r = 0; r < 8; ++r) acc[r] = (v8f){0.f, 0.f, 0.f, 0.f, 0.f, 0.f, 0.f, 0.f};

  for (int c = 0; c < CSUB; ++c) {
    // B operands for all 4 k-chunks of this subspace (b64 each, coalesced)
    v2f bq[4];
#pragma unroll
    for (int q = 0; q < 4; ++q)
      bq[q] = P[(size_t)(c * 8 + q * 2 + half) * MOUT + m0 + l15];

    // codes for the 8 row-tiles (16B broadcast segment per tile)
    unsigned int kr[8];
#pragma unroll
    for (int r = 0; r < 8; ++r)
      kr[r] = codesT[(size_t)c * NROWS + n0 + r * 16 + l15];

#pragma unroll
    for (int q = 0; q < 4; ++q) {
      const unsigned int k0 = (unsigned int)(q * 4 + half * 2);
#pragma unroll
      for (int r = 0; r < 8; ++r) {
        // one-hot A operand in the ISA 16x4 f32 A layout:
        // VGPR0: K = 4q + 2*half, VGPR1: K = 4q + 2*half + 1, M = lane&15
        v2f a;
        a[0] = (kr[r] == k0)     ? 1.0f : 0.0f;
        a[1] = (kr[r] == k0 + 1) ? 1.0f : 0.0f;
        acc[r] = __builtin_amdgcn_wmma_f32_16x16x4_f32(
            false, a, false, bq[q], (short)0, acc[r], false, false);
      }
    }
  }

  // D layout: VGPR j -> row j (lanes 0-15) / row j+8 (lanes 16-31), col = lane&15
#pragma unroll
  for (int r = 0; r < 8; ++r) {
#pragma unroll
    for (int j = 0; j < 8; ++j) {
      out[(size_t)(n0 + r * 16 + half * 8 + j) * MOUT + m0 + l15] = acc[r][j];
    }
  }
}

// ---------------------------------------------------------------------------
extern "C" void kernel_launch(void* const* d_in, const int* in_sizes, int n_in,
                              void* d_out, int out_size, void* d_ws, size_t ws_size,
                              hipStream_t stream) {
  const float* I = (const float*)d_in[0];   // (N, 512)
  const float* A = (const float*)d_in[1];   // (64, 8, 4)
  const float* T = (const float*)d_in[2];   // (960,)
  const float* L = (const float*)d_in[3];   // (512, 64, 16)
  // d_in[4] = S, d_in[5] = B: structure is hardcoded (selection/bit matrices)

  unsigned char* codesT = (unsigned char*)d_ws;                    // 1 MB
  v2f* P = (v2f*)((char*)d_ws + (size_t)CSUB * NROWS);             // 2 MB
  float* out = (float*)d_out;

  halut_encode<<<NROWS / 4, 256, 0, stream>>>(I, A, T, codesT);
  halut_pack_lut<<<(CSUB * 8 * MOUT) / 256, 256, 0, stream>>>(L, P);
  // 128 n-groups x 32 m-tiles = 4096 waves = 512 blocks of 8 waves
  halut_decode<<<512, 256, 0, stream>>>(P, codesT, out);
}